// Decoder_Model_EBV_66984309949054
// MI455X (gfx1250) — compile-verified
//
#include <hip/hip_runtime.h>

typedef float v2f __attribute__((ext_vector_type(2)));
typedef float v8f __attribute__((ext_vector_type(8)));

#define EMBED   512
#define BASIS   256
#define KCHUNK  128
#define LDSPAD  132   // 132*4B = 528B row stride: 16B aligned, stride%64banks==4 -> conflict-free

// ---------------------------------------------------------------------------
// Kernel 1: X_trans[n][b] = sum_k X[n][k] * W[b][k]   via V_WMMA_F32_16X16X4_F32
// Block: 256 threads = 8 waves. Block covers 32 rows x 256 cols.
//   wave w: strip = w>>2 (16-row strip), cg = w&3 (64-col group, 4 accum tiles)
// ---------------------------------------------------------------------------
__global__ __launch_bounds__(256)
void gemm_xtrans(const float* __restrict__ X, const float* __restrict__ W,
                 float* __restrict__ Xt) {
  __shared__ float tile[32][LDSPAD];

  const int tid   = threadIdx.x;
  const int lane  = tid & 31;
  const int wave  = tid >> 5;
  const int strip = wave >> 2;        // 0..1
  const int cg    = wave & 3;         // 0..3
  const int row0  = blockIdx.x * 32;
  const int lhalf = lane >> 4;        // 0 | 1  (lanes 16-31 carry K+2 per 16x4 A layout)
  const int lmod  = lane & 15;

  v8f acc0 = {}, acc1 = {}, acc2 = {}, acc3 = {};

  for (int kc = 0; kc < EMBED; kc += KCHUNK) {
    __syncthreads();
    // Stage 32 x 128 floats of X into LDS: 1024 float4s / 256 threads = 4 each
#pragma unroll
    for (int i = 0; i < 4; ++i) {
      int f4 = tid + i * 256;         // 0..1023 ; 32 float4 per row
      int r  = f4 >> 5;
      int c4 = f4 & 31;
      float4 v = *(const float4*)(X + (size_t)(row0 + r) * EMBED + kc + c4 * 4);
      *(float4*)&tile[r][c4 * 4] = v;
    }
    __syncthreads();

#pragma unroll 4
    for (int kk = 0; kk < KCHUNK; kk += 4) {
      // A fragment: 16x4 f32; lane l: row = l&15, K = kk + 2*(l>>4) .. +1
      v2f a = *(const v2f*)&tile[strip * 16 + lmod][kk + 2 * lhalf];
      // B fragment: B[k][b] = W[b][k]; lane l: col b = cg*64+t*16+(l&15),
      // K = kc+kk+2*(l>>4) -> adjacent pair in row-major W => one b64 load
      const float* wb = W + (size_t)(cg * 64 + lmod) * EMBED + kc + kk + 2 * lhalf;
      v2f b0 = *(const v2f*)(wb);
      v2f b1 = *(const v2f*)(wb + 16 * EMBED);
      v2f b2 = *(const v2f*)(wb + 32 * EMBED);
      v2f b3 = *(const v2f*)(wb + 48 * EMBED);
      acc0 = __builtin_amdgcn_wmma_f32_16x16x4_f32(false, a, false, b0, (short)0, acc0, false, false);
      acc1 = __builtin_amdgcn_wmma_f32_16x16x4_f32(false, a, false, b1, (short)0, acc1, false, false);
      acc2 = __builtin_amdgcn_wmma_f32_16x16x4_f32(false, a, false, b2, (short)0, acc2, false, false);
      acc3 = __builtin_amdgcn_wmma_f32_16x16x4_f32(false, a, false, b3, (short)0, acc3, false, false);
    }
  }

  // D layout: VGPR j -> M = j (lanes 0-15), M = j+8 (lanes 16-31); N = lane&15
  const int bbase = cg * 64 + lmod;
#pragma unroll
  for (int j = 0; j < 8; ++j) {
    size_t roff = (size_t)(row0 + strip * 16 + j + 8 * lhalf) * BASIS;
    Xt[roff + bbase +  0] = acc0[j];
    Xt[roff + bbase + 16] = acc1[j];
    Xt[roff + bbase + 32] = acc2[j];
    Xt[roff + bbase + 48] = acc3[j];
  }
}

// ---------------------------------------------------------------------------
// Kernel 2: scores[e] = dot(Xt[src[e]] - Xt[tgt[e]], ebvecs[type[e]])
// One wave32 per edge; gathers are L2-hot (Xt = 102.4MB < 192MB L2).
// ---------------------------------------------------------------------------
__global__ __launch_bounds__(256)
void edge_score(const float* __restrict__ Xt, const int* __restrict__ src,
                const int* __restrict__ tgt, const int* __restrict__ rel,
                const float* __restrict__ ebv, float* __restrict__ out,
                int nedges) {
  int e    = (int)((blockIdx.x * blockDim.x + threadIdx.x) >> 5);
  int lane = threadIdx.x & 31;
  if (e >= nedges) return;

  int s = src[e], t = tgt[e], r = rel[e];
  const float4* ps = (const float4*)(Xt  + (size_t)s * BASIS);
  const float4* pt = (const float4*)(Xt  + (size_t)t * BASIS);
  const float4* pe = (const float4*)(ebv + (size_t)r * BASIS);

  float acc = 0.f;
#pragma unroll
  for (int j = 0; j < 2; ++j) {
    float4 a = ps[lane + 32 * j];
    float4 b = pt[lane + 32 * j];
    float4 v = pe[lane + 32 * j];
    acc += (a.x - b.x) * v.x + (a.y - b.y) * v.y +
           (a.z - b.z) * v.z + (a.w - b.w) * v.w;
  }
#pragma unroll
  for (int off = 16; off > 0; off >>= 1)
    acc += __shfl_down(acc, off, 32);
  if (lane == 0) out[e] = acc;
}

extern "C" void kernel_launch(void* const* d_in, const int* in_sizes, int n_in,
                              void* d_out, int out_size, void* d_ws, size_t ws_size,
                              hipStream_t stream) {
  const float* X     = (const float*)d_in[0];
  const int*   edges = (const int*)d_in[1];   // (2, E) int32 (JAX x64-off)
  const int*   etype = (const int*)d_in[2];   // (1, E)
  const float* W     = (const float*)d_in[3]; // (256, 512)
  const float* ebv   = (const float*)d_in[4]; // (500, 256)
  float*       out   = (float*)d_out;         // (1, E)
  float*       Xt    = (float*)d_ws;          // (N, 256) scratch = 102.4 MB

  int nnodes = in_sizes[0] / EMBED;           // 100000
  int nedges = in_sizes[2];                   // 300000

  gemm_xtrans<<<nnodes / 32, 256, 0, stream>>>(X, W, Xt);

  int nblk = (nedges + 7) / 8;                // 8 waves (edges) per block
  edge_score<<<nblk, 256, 0, stream>>>(Xt, edges, edges + nedges, etype, ebv,
                                       out, nedges);
}